// HistogramColorLoss_50483045597472
// MI455X (gfx1250) — compile-verified
//
#include <hip/hip_runtime.h>
#include <hip/hip_bf16.h>

#define BINS 256
#define NCH 3
#define HIST (NCH * BINS)            // 768 per tensor
#define WAVES_PER_BLOCK 8            // 256 threads / wave32
#define PLANE_VEC_SHIFT 18           // H*W = 2^20 floats = 2^18 float4s

typedef __attribute__((ext_vector_type(2))) float v2f;
typedef __attribute__((ext_vector_type(8))) float v8f;

// Order-preserving float->uint key (works for any sign).
__device__ __forceinline__ unsigned flipKey(float f) {
    unsigned u = __float_as_uint(f);
    return (u & 0x80000000u) ? ~u : (u | 0x80000000u);
}
__device__ __forceinline__ float unflipKey(unsigned k) {
    return __uint_as_float((k & 0x80000000u) ? (k ^ 0x80000000u) : ~k);
}

// Workspace layout (uint):
//   w[0]=min_in key  w[1]=max_in key  w[2]=min_tg key  w[3]=max_tg key
//   w[16 .. 16+768)        hist_in
//   w[16+768 .. 16+1536)   hist_tg
#define WS_HIST_OFF 16

__global__ void initWS(unsigned* __restrict__ w) {
    int t = blockIdx.x * blockDim.x + threadIdx.x;
    if (t == 0) {
        w[0] = 0xFFFFFFFFu;  // min slots: max key
        w[1] = 0u;           // max slots: min key
        w[2] = 0xFFFFFFFFu;
        w[3] = 0u;
    }
    int n = 2 * HIST;
    for (int i = t; i < n; i += blockDim.x * gridDim.x) w[WS_HIST_OFF + i] = 0u;
}

__global__ __launch_bounds__(256)
void minmaxKernel(const float4* __restrict__ in, const float4* __restrict__ tg,
                  long long nvec, unsigned* __restrict__ w) {
    long long stride = (long long)gridDim.x * blockDim.x;
    long long i = (long long)blockIdx.x * blockDim.x + threadIdx.x;
    float mni = INFINITY, mxi = -INFINITY, mnt = INFINITY, mxt = -INFINITY;
    for (; i < nvec; i += stride) {
        float4 a = in[i];
        float4 b = tg[i];
        if (i + stride < nvec) {
            __builtin_prefetch(&in[i + stride], 0, 1);   // global_prefetch_b8
            __builtin_prefetch(&tg[i + stride], 0, 1);
        }
        mni = fminf(mni, fminf(fminf(a.x, a.y), fminf(a.z, a.w)));
        mxi = fmaxf(mxi, fmaxf(fmaxf(a.x, a.y), fmaxf(a.z, a.w)));
        mnt = fminf(mnt, fminf(fminf(b.x, b.y), fminf(b.z, b.w)));
        mxt = fmaxf(mxt, fmaxf(fmaxf(b.x, b.y), fmaxf(b.z, b.w)));
    }
    // wave32 reduction
    for (int off = 16; off; off >>= 1) {
        mni = fminf(mni, __shfl_down(mni, off));
        mxi = fmaxf(mxi, __shfl_down(mxi, off));
        mnt = fminf(mnt, __shfl_down(mnt, off));
        mxt = fmaxf(mxt, __shfl_down(mxt, off));
    }
    if ((threadIdx.x & 31) == 0) {
        atomicMin(&w[0], flipKey(mni));
        atomicMax(&w[1], flipKey(mxi));
        atomicMin(&w[2], flipKey(mnt));
        atomicMax(&w[3], flipKey(mxt));
    }
}

__device__ __forceinline__ int binOf(float x, float mn, float scale) {
    int j = (int)floorf((x - mn) * scale);
    return j < 0 ? 0 : (j > BINS - 1 ? BINS - 1 : j);
}

__global__ __launch_bounds__(256)
void histKernel(const float4* __restrict__ in, const float4* __restrict__ tg,
                long long nvec, unsigned* __restrict__ w) {
    // Per-wave private histograms: no inter-wave LDS contention.
    __shared__ unsigned lh[WAVES_PER_BLOCK * 2 * HIST];   // 48 KB
    int tid = threadIdx.x;
    for (int i = tid; i < WAVES_PER_BLOCK * 2 * HIST; i += 256) lh[i] = 0u;
    __syncthreads();

    float mni = unflipKey(w[0]), mxi = unflipKey(w[1]);
    float mnt = unflipKey(w[2]), mxt = unflipKey(w[3]);
    float si = (float)BINS / (mxi - mni);
    float st = (float)BINS / (mxt - mnt);

    unsigned* hin = &lh[(tid >> 5) * 2 * HIST];
    unsigned* htg = hin + HIST;

    long long stride = (long long)gridDim.x * blockDim.x;
    for (long long v = (long long)blockIdx.x * blockDim.x + tid; v < nvec; v += stride) {
        float4 a = in[v];
        float4 b = tg[v];
        if (v + stride < nvec) {
            __builtin_prefetch(&in[v + stride], 0, 1);
            __builtin_prefetch(&tg[v + stride], 0, 1);
        }
        // All 4 elements of a float4 share a channel (plane is 2^18 float4s).
        int ch = (int)((unsigned)(v >> PLANE_VEC_SHIFT) % NCH);
        unsigned* hi = hin + ch * BINS;
        unsigned* ht = htg + ch * BINS;
        atomicAdd(&hi[binOf(a.x, mni, si)], 1u);   // ds_add_u32
        atomicAdd(&hi[binOf(a.y, mni, si)], 1u);
        atomicAdd(&hi[binOf(a.z, mni, si)], 1u);
        atomicAdd(&hi[binOf(a.w, mni, si)], 1u);
        atomicAdd(&ht[binOf(b.x, mnt, st)], 1u);
        atomicAdd(&ht[binOf(b.y, mnt, st)], 1u);
        atomicAdd(&ht[binOf(b.z, mnt, st)], 1u);
        atomicAdd(&ht[binOf(b.w, mnt, st)], 1u);
    }
    __syncthreads();

    // Merge the 8 wave-private copies, one global atomic per bin per block.
    for (int i = tid; i < 2 * HIST; i += 256) {
        unsigned s = 0;
        #pragma unroll
        for (int wv = 0; wv < WAVES_PER_BLOCK; ++wv) s += lh[wv * 2 * HIST + i];
        if (s) atomicAdd(&w[WS_HIST_OFF + i], s);
    }
}

// One wave. Computes sum((h_in - h_tg)^2) over 768 bins with f32 WMMA:
// with the 16x4 f32 A layout and 4x16 f32 B layout, passing the same
// registers as A and B yields B == A^T, so diag(D) = row sums of squares.
__global__ __launch_bounds__(32)
void finalKernel(const unsigned* __restrict__ w, float* __restrict__ out) {
    const unsigned* hin = w + WS_HIST_OFF;
    const unsigned* htg = hin + HIST;
    int lane = threadIdx.x;
    int m  = lane & 15;
    int k0 = (lane >> 4) << 1;   // lanes 0-15 hold K=0,1 ; lanes 16-31 hold K=2,3

    v8f c = {};
    #pragma unroll
    for (int chunk = 0; chunk < HIST / 64; ++chunk) {   // 12 chunks of 64
        int base = chunk * 64 + m * 4 + k0;
        v2f a;
        a.x = (float)hin[base]     - (float)htg[base];
        a.y = (float)hin[base + 1] - (float)htg[base + 1];
        c = __builtin_amdgcn_wmma_f32_16x16x4_f32(
                /*neg_a=*/false, a, /*neg_b=*/false, a,
                /*c_mod=*/(short)0, c, /*reuse_a=*/false, /*reuse_b=*/false);
    }

    // Diagonal of 16x16 f32 D: D[i][i] for i<8 -> lane i, component i;
    // for i>=8 -> lane i+16 (i.e. lanes 24..31), component i-8.
    float diag = 0.0f;
    if (lane < 8)        diag = c[lane];
    else if (lane >= 24) diag = c[lane - 24];

    for (int off = 16; off; off >>= 1) diag += __shfl_down(diag, off);

    if (lane == 0) {
        float loss = diag / (float)HIST;     // mean over C*BINS
        float r = 1.0f - 1.0f / loss;
        if (isinf(r)) r = 1.0f;              // reference maps any inf -> 1.0
        out[0] = r;
    }
}

extern "C" void kernel_launch(void* const* d_in, const int* in_sizes, int n_in,
                              void* d_out, int out_size, void* d_ws, size_t ws_size,
                              hipStream_t stream) {
    const float4* in = (const float4*)d_in[0];
    const float4* tg = (const float4*)d_in[1];
    unsigned* w = (unsigned*)d_ws;
    float* out = (float*)d_out;

    long long total = (long long)in_sizes[0];   // 16*3*1024*1024
    long long nvec  = total >> 2;               // float4 count

    initWS<<<8, 256, 0, stream>>>(w);
    minmaxKernel<<<2048, 256, 0, stream>>>(in, tg, nvec, w);
    histKernel<<<2048, 256, 0, stream>>>(in, tg, nvec, w);
    finalKernel<<<1, 32, 0, stream>>>(w, out);
}